// ReactDiffSolver_55628416418501
// MI455X (gfx1250) — compile-verified
//
#include <hip/hip_runtime.h>

// Reaction-diffusion forward-Euler on gfx1250.
// - State (2x128x128 f32 = 128 KB/batch) lives in LDS for all 1000 substeps;
//   each thread's own 16-column segment additionally lives in registers,
//   updated in place (LDS mirror exists only for neighbor threads + TDM).
// - TDM stages the initial state (tensor_load_to_lds) and streams every
//   128 KB snapshot to HBM (tensor_store_from_lds), overlapped with the
//   next substep's compute phase (which only reads LDS).
// out shape: (16, 51, 2, 128, 128) f32.

#define NX          128
#define NPTS        (NX * NX)           // 16384 per field
#define STATE_ELEMS (2 * NPTS)          // 32768 floats / batch = 128 KB
#define NBATCH      16
#define NSNAP       51
#define SUBSTEPS    20
#define NINTERVAL   50
#define HSTEP       0.001f
#define DXV         0.06451612903f

typedef unsigned int u32x4 __attribute__((ext_vector_type(4)));
typedef int          i32x4 __attribute__((ext_vector_type(4)));
typedef int          i32x8 __attribute__((ext_vector_type(8)));

#if __has_builtin(__builtin_amdgcn_tensor_load_to_lds) && \
    __has_builtin(__builtin_amdgcn_tensor_store_from_lds) && \
    __has_builtin(__builtin_amdgcn_s_wait_tensorcnt)
#define HAVE_TDM 1
#else
#define HAVE_TDM 0
#endif

__device__ __forceinline__ void load16(float* r, const float* l) {
#pragma unroll
  for (int q = 0; q < 4; ++q) {
    float4 t = ((const float4*)l)[q];
    r[4*q+0] = t.x; r[4*q+1] = t.y; r[4*q+2] = t.z; r[4*q+3] = t.w;
  }
}

__device__ __forceinline__ void copy16(float* g, const float* l) {
#pragma unroll
  for (int q = 0; q < 4; ++q)
    ((float4*)g)[q] = ((const float4*)l)[q];
}

#if HAVE_TDM
// D# for one contiguous row of STATE_ELEMS f32 elements (128 KB tile).
__device__ __forceinline__ void make_desc(unsigned long long ga, unsigned ldsOff,
                                          u32x4& g0, i32x8& g1) {
  g0[0] = 1u;                                              // count=1, user mode
  g0[1] = ldsOff;                                          // lds_addr (bytes)
  g0[2] = (unsigned)(ga & 0xFFFFFFFFull);                  // global_addr[31:0]
  g0[3] = (unsigned)((ga >> 32) & 0x1FFFFFFull)            // global_addr[56:32]
        | (2u << 30);                                      // type = 2 ("image")
  g1[0] = (int)(2u << 16);                                 // data_size = 4 B
  g1[1] = (int)(((unsigned)STATE_ELEMS & 0xFFFFu) << 16);  // tensor_dim0 lo16
  g1[2] = (int)((((unsigned)STATE_ELEMS >> 16) & 0xFFFFu)  // tensor_dim0 hi16
                | (1u << 16));                             // tensor_dim1 = 1
  g1[3] = (int)(((unsigned)STATE_ELEMS & 0xFFFFu) << 16);  // tile_dim0 = 32768
  g1[4] = 1;                                               // tile_dim1 = 1
  g1[5] = (int)STATE_ELEMS;                                // dim0_stride lo32
  g1[6] = (int)(((unsigned)STATE_ELEMS & 0xFFFFu) << 16);  // dim1_stride lo16
  g1[7] = 0;
}

__device__ __forceinline__ void tdm_load(const void* g, void* l) {
  u32x4 g0; i32x8 g1;
  make_desc((unsigned long long)g, (unsigned)(unsigned long long)l, g0, g1);
  i32x4 z4 = {0, 0, 0, 0};
#if defined(__clang_major__) && (__clang_major__ >= 23)
  i32x8 z8 = {0, 0, 0, 0, 0, 0, 0, 0};
  __builtin_amdgcn_tensor_load_to_lds(g0, g1, z4, z4, z8, 0);
#else
  __builtin_amdgcn_tensor_load_to_lds(g0, g1, z4, z4, 0);
#endif
}

__device__ __forceinline__ void tdm_store(void* g, const void* l) {
  u32x4 g0; i32x8 g1;
  make_desc((unsigned long long)g, (unsigned)(unsigned long long)l, g0, g1);
  i32x4 z4 = {0, 0, 0, 0};
#if defined(__clang_major__) && (__clang_major__ >= 23)
  i32x8 z8 = {0, 0, 0, 0, 0, 0, 0, 0};
  __builtin_amdgcn_tensor_store_from_lds(g0, g1, z4, z4, z8, 0);
#else
  __builtin_amdgcn_tensor_store_from_lds(g0, g1, z4, z4, 0);
#endif
}
#endif  // HAVE_TDM

__global__ __launch_bounds__(1024, 1)
void rd_solver(const float* __restrict__ init,
               const float* __restrict__ params,
               float* __restrict__ out)
{
  extern __shared__ float smem[];       // [2][128][128] = 128 KB (of 320 KB/WGP)
  float* Us = smem;
  float* Vs = smem + NPTS;

  const int b   = blockIdx.x;
  const int tid = threadIdx.x;
  const int r   = tid >> 3;             // row 0..127
  const int c0  = (tid & 7) << 4;       // col start: 0,16,...,112 (64B aligned)

  const float a       = params[b*3 + 0];
  const float bb      = params[b*3 + 1];
  const float k       = params[b*3 + 2];
  const float inv_dx2 = 1.0f / (DXV * DXV);

  const float* gsrc = init + (size_t)b * STATE_ELEMS;
  float* outB = out + (size_t)b * NSNAP * STATE_ELEMS;

  const int rowm = ((r + NX - 1) & (NX - 1)) * NX;
  const int row0 = r * NX;
  const int rowp = ((r + 1) & (NX - 1)) * NX;
  const int cm   = row0 + ((c0 + NX - 1) & (NX - 1));
  const int cp   = row0 + ((c0 + 16) & (NX - 1));
  const int base = row0 + c0;

  // ---- Stage initial state into LDS, then kick off the t=0 snapshot DMA ----
#if HAVE_TDM
  if (tid < 32) {                       // wave 0 issues the DMA descriptors
    tdm_load(gsrc, smem);
    __builtin_amdgcn_s_wait_tensorcnt(0);
  }
  __syncthreads();                      // LDS state visible to all waves
  if (tid < 32)
    tdm_store(outB, smem);              // t=0 snapshot; overlaps substep-0 reads
#else
  for (int p = tid; p < STATE_ELEMS; p += (int)blockDim.x)
    smem[p] = gsrc[p];
  __syncthreads();
  copy16(outB + base,        Us + base);
  copy16(outB + NPTS + base, Vs + base);
#endif

  // This thread's own 16-column segment, carried in registers across all
  // substeps (this thread is its sole writer; LDS mirror feeds neighbors/TDM).
  float uc[16], vc[16];
  load16(uc, Us + base);
  load16(vc, Vs + base);

  for (int it = 0; it < NINTERVAL; ++it) {
    for (int ss = 0; ss < SUBSTEPS; ++ss) {
      // ---- compute phase: only READS LDS (safe vs in-flight TDM store) ----
      float ut[16], ub[16], vt[16], vb[16];
      load16(ut, Us + rowm + c0);
      load16(ub, Us + rowp + c0);
      load16(vt, Vs + rowm + c0);
      load16(vb, Vs + rowp + c0);
      const float uL = Us[cm], uR = Us[cp];
      const float vL = Vs[cm], vR = Vs[cp];

      // In-place register update: at iteration j, uc[j+1]/vc[j+1] still hold
      // old values; old left neighbor is carried in upL/vpL. No copy loop.
      float upL = uL, vpL = vL;
#pragma unroll
      for (int j = 0; j < 16; ++j) {
        const float u  = uc[j];
        const float v  = vc[j];
        const float ur = (j == 15) ? uR : uc[j+1];
        const float vr = (j == 15) ? vR : vc[j+1];
        const float lapU = (ut[j] + ub[j] + upL + ur - 4.0f * u) * inv_dx2;
        const float lapV = (vt[j] + vb[j] + vpL + vr - 4.0f * v) * inv_dx2;
        const float dU = fmaf(a,  lapU, u - u*u*u - k - v);
        const float dV = fmaf(bb, lapV, u - v);
        uc[j] = fmaf(HSTEP, dU, u);
        vc[j] = fmaf(HSTEP, dV, v);
        upL = u;                        // old value becomes next left neighbor
        vpL = v;
      }

#if HAVE_TDM
      // Before the first in-place LDS write of this interval, make sure the
      // previous snapshot DMA (which reads LDS) has drained.
      if (ss == 0 && tid < 32)
        __builtin_amdgcn_s_wait_tensorcnt(0);
#endif
      __syncthreads();                  // all reads (and snapshot DMA) done
#pragma unroll
      for (int q = 0; q < 4; ++q) {
        ((float4*)(Us + base))[q] =
            make_float4(uc[4*q], uc[4*q+1], uc[4*q+2], uc[4*q+3]);
        ((float4*)(Vs + base))[q] =
            make_float4(vc[4*q], vc[4*q+1], vc[4*q+2], vc[4*q+3]);
      }
      __syncthreads();                  // updated state visible to all waves
    }

    // ---- snapshot t = it+1: one 128 KB TDM DMA, overlapped with next interval ----
#if HAVE_TDM
    if (tid < 32)
      tdm_store(outB + (size_t)(it + 1) * STATE_ELEMS, smem);
#else
    float* dst = outB + (size_t)(it + 1) * STATE_ELEMS;
    copy16(dst + base,        Us + base);
    copy16(dst + NPTS + base, Vs + base);
#endif
  }

#if HAVE_TDM
  if (tid < 32)
    __builtin_amdgcn_s_wait_tensorcnt(0);  // drain final snapshot DMA
#endif
}

extern "C" void kernel_launch(void* const* d_in, const int* in_sizes, int n_in,
                              void* d_out, int out_size, void* d_ws, size_t ws_size,
                              hipStream_t stream) {
  const float* init   = (const float*)d_in[0];   // (16, 2, 128, 128) f32
  const float* params = (const float*)d_in[1];   // (16, 3) f32
  float* out = (float*)d_out;                    // (16, 51, 2, 128, 128) f32

  const size_t shmem = (size_t)STATE_ELEMS * sizeof(float);  // 128 KB dynamic LDS
  rd_solver<<<dim3(NBATCH), dim3(1024), shmem, stream>>>(init, params, out);

  (void)in_sizes; (void)n_in; (void)out_size; (void)d_ws; (void)ws_size;
}